// CCFLoss_38414187495653
// MI455X (gfx1250) — compile-verified
//
#include <hip/hip_runtime.h>
#include <hip/hip_bf16.h>

// Problem geometry (fixed by the reference).
#define BB 16
#define CC 11
#define HH 256
#define WW 256
#define HWN (HH * WW)            // 65536 elements per (b,c) plane
#define NPLANES (BB * CC)        // 176
#define N_TOTAL ((double)BB * CC * HH * WW)
#define CHUNKS 16                // chunks per plane
#define CHUNK_ELEMS (HWN / CHUNKS) // 4096
#define TPB 256
#define E_ITERS (CHUNK_ELEMS / (TPB * 4)) // 4 float4s per thread
#define N_V_CH 5

typedef float v4f __attribute__((ext_vector_type(4)));
typedef float v2f __attribute__((ext_vector_type(2)));
typedef float v8f __attribute__((ext_vector_type(8)));

// Monotonic float -> uint key (total order matching float compare).
__device__ __forceinline__ unsigned int fkey(float f) {
    unsigned int b = __float_as_uint(f);
    return b ^ (unsigned int)(((int)b >> 31) | 0x80000000u);
}

__global__ void init_ws(double* __restrict__ sums,
                        unsigned long long* __restrict__ amax) {
    int t = threadIdx.x;
    if (t < 3) sums[t] = 0.0;
    if (t < NPLANES) amax[t] = 0ULL;
}

__global__ __launch_bounds__(TPB) void main_pass(
    const float* __restrict__ hp, const float* __restrict__ yp,
    const float* __restrict__ xp, const float* __restrict__ ht,
    const float* __restrict__ yt, const float* __restrict__ xt,
    const float* __restrict__ mk,
    double* __restrict__ sums, unsigned long long* __restrict__ amax) {
    const int plane = blockIdx.x / CHUNKS;
    const int chunk = blockIdx.x % CHUNKS;
    const int tid   = threadIdx.x;
    const long long base = (long long)plane * HWN + (long long)chunk * CHUNK_ELEMS;

    float hs = 0.f, ys = 0.f, xs = 0.f;
    unsigned long long best = 0ULL;

#pragma unroll
    for (int e = 0; e < E_ITERS; ++e) {
        const int off = (e * TPB + tid) * 4;          // 16B aligned
        const long long g = base + off;
        // Streamed once; 323 MB total > 192 MB L2 -> non-temporal 128-bit loads.
        v4f a_hp = __builtin_nontemporal_load((const v4f*)(hp + g));
        v4f a_ht = __builtin_nontemporal_load((const v4f*)(ht + g));
        v4f a_mk = __builtin_nontemporal_load((const v4f*)(mk + g));
        v4f a_yp = __builtin_nontemporal_load((const v4f*)(yp + g));
        v4f a_yt = __builtin_nontemporal_load((const v4f*)(yt + g));
        v4f a_xp = __builtin_nontemporal_load((const v4f*)(xp + g));
        v4f a_xt = __builtin_nontemporal_load((const v4f*)(xt + g));
        const int lidx = chunk * CHUNK_ELEMS + off;    // flat index within plane
#pragma unroll
        for (int j = 0; j < 4; ++j) {
            float m = a_mk[j], h = a_ht[j];
            float dh = a_hp[j] * m - h * m;
            float dy = a_yp[j] * h - a_yt[j] * h;
            float dx = a_xp[j] * h - a_xt[j] * h;
            hs = fmaf(dh, dh, hs);
            ys = fmaf(dy, dy, ys);
            xs = fmaf(dx, dx, xs);
            // argmax: key in high 32 bits, ~idx low (first occurrence wins ties)
            unsigned long long cand =
                ((unsigned long long)fkey(h) << 32) |
                (unsigned long long)(~(unsigned int)(lidx + j));
            best = cand > best ? cand : best;
        }
    }

    // wave32 reductions
#pragma unroll
    for (int o = 16; o > 0; o >>= 1) {
        hs += __shfl_xor(hs, o, 32);
        ys += __shfl_xor(ys, o, 32);
        xs += __shfl_xor(xs, o, 32);
        unsigned long long ob = __shfl_xor(best, o, 32);
        best = ob > best ? ob : best;
    }

    __shared__ float s3[3][8];
    __shared__ unsigned long long sm[8];
    const int wave = tid >> 5, lane = tid & 31;
    if (lane == 0) {
        s3[0][wave] = hs; s3[1][wave] = ys; s3[2][wave] = xs; sm[wave] = best;
    }
    __syncthreads();

    if (tid < 32) {
        // Cross-wave sum via one V_WMMA_F32_16X16X4_F32:
        // A(16x4) rows hold groups of 4 wave-partials, B = ones -> D rows = row-sums.
        //   rows 0,1 : heat partials (waves 0-3 / 4-7)
        //   rows 2,3 : offy partials
        //   rows 4,5 : offx partials
        // A layout (f32 16x4): lane m (0-15) holds K=0,1 in v0,v1; lane m+16 holds K=2,3.
        const int row = tid & 15, hi = tid >> 4;
        v2f a = {0.f, 0.f};
        if (row < 6) {
            const int q  = row >> 1;
            const int wb = ((row & 1) << 2) | (hi << 1);
            a.x = s3[q][wb];
            a.y = s3[q][wb + 1];
        }
        v2f bones = {1.f, 1.f};
        v8f c = {};
        c = __builtin_amdgcn_wmma_f32_16x16x4_f32(
            /*neg_a=*/false, a, /*neg_b=*/false, bones,
            /*c_mod=*/(short)0, c, /*reuse_a=*/false, /*reuse_b=*/false);
        if (tid == 0) {
            // lane 0 holds column N=0: c[r] = D[r][0] = sum of A row r
            atomicAdd(&sums[0], (double)c[0] + (double)c[1]);
            atomicAdd(&sums[1], (double)c[2] + (double)c[3]);
            atomicAdd(&sums[2], (double)c[4] + (double)c[5]);
            unsigned long long m = sm[0];
#pragma unroll
            for (int i = 1; i < 8; ++i) m = sm[i] > m ? sm[i] : m;
            atomicMax(&amax[plane], m);
        }
    }
}

__device__ __forceinline__ float softplusf(float x) {
    return fmaxf(x, 0.f) + log1pf(expf(-fabsf(x)));
}

__device__ __forceinline__ double en_term(double s) {
    double beta = (s - 1.0) / s;
    return (1.0 - pow(beta, s)) / (1.0 - beta);
}

__global__ __launch_bounds__(256) void final_pass(
    const float* __restrict__ cp, const float* __restrict__ ct,
    const double* __restrict__ sums, const unsigned long long* __restrict__ amax,
    const float* __restrict__ vw, const float* __restrict__ dw,
    float* __restrict__ out) {
    const int t = threadIdx.x;
    float svl = 0.f, svm = 0.f, sdl = 0.f, sdm = 0.f;

    if (t < NPLANES) {
        // class-balanced pos_weights, computed in double to match numpy
        double w0v = 1.0 / (en_term(8000.0) + 1e-5);
        double w1v = 1.0 / (en_term(2000.0) + 1e-5);
        float POSWV = (float)(w1v / (w0v + 1e-5));
        double w0d = 1.0 / (en_term(7000.0) + 1e-5);
        double w1d = 1.0 / (en_term(3000.0) + 1e-5);
        float POSWD = (float)(w1d / (w0d + 1e-5));

        const int c = t % CC;
        unsigned int idx = ~(unsigned int)(amax[t] & 0xFFFFFFFFull);
        const long long g = (long long)t * HWN + idx;
        float x = cp[g];
        float y = ct[g];
        bool valid = (y >= 0.f);
        float spn = softplusf(-x);
        float spp = softplusf(x);
        if (valid && (c < N_V_CH)) {
            svl = POSWV * y * spn + (1.f - y) * spp;
            svm = 1.f;
        }
        if (valid && (c >= N_V_CH)) {
            float yd = (y >= 1.f) ? 1.f : 0.f;
            sdl = POSWD * yd * spn + (1.f - yd) * spp;
            sdm = 1.f;
        }
    }

#pragma unroll
    for (int o = 16; o > 0; o >>= 1) {
        svl += __shfl_xor(svl, o, 32);
        svm += __shfl_xor(svm, o, 32);
        sdl += __shfl_xor(sdl, o, 32);
        sdm += __shfl_xor(sdm, o, 32);
    }
    __shared__ float r4[4][8];
    const int wave = t >> 5, lane = t & 31;
    if (lane == 0) { r4[0][wave] = svl; r4[1][wave] = svm; r4[2][wave] = sdl; r4[3][wave] = sdm; }
    __syncthreads();

    if (t == 0) {
        float SVL = 0.f, SVM = 0.f, SDL = 0.f, SDM = 0.f;
#pragma unroll
        for (int i = 0; i < 8; ++i) {
            SVL += r4[0][i]; SVM += r4[1][i]; SDL += r4[2][i]; SDM += r4[3][i];
        }
        const double inv_n = 1.0 / N_TOTAL;
        float heat = (float)(sums[0] * inv_n);
        float offy = (float)(sums[1] * inv_n);
        float offx = (float)(sums[2] * inv_n);
        float vcls = SVL / fmaxf(SVM, 1.f);
        float dcls = SDL / fmaxf(SDM, 1.f);
        out[0] = heat + offy + offx + vcls * vw[0] + dcls * dw[0];
    }
}

extern "C" void kernel_launch(void* const* d_in, const int* in_sizes, int n_in,
                              void* d_out, int out_size, void* d_ws, size_t ws_size,
                              hipStream_t stream) {
    (void)in_sizes; (void)n_in; (void)out_size; (void)ws_size;
    const float* heat_p = (const float*)d_in[0];
    const float* offy_p = (const float*)d_in[1];
    const float* offx_p = (const float*)d_in[2];
    const float* clss_p = (const float*)d_in[3];
    const float* heat_t = (const float*)d_in[4];
    const float* offy_t = (const float*)d_in[5];
    const float* offx_t = (const float*)d_in[6];
    const float* clss_t = (const float*)d_in[7];
    const float* masks  = (const float*)d_in[8];
    const float* vw     = (const float*)d_in[9];
    const float* dw     = (const float*)d_in[10];

    double* sums = (double*)d_ws;                                   // 3 doubles
    unsigned long long* amax = (unsigned long long*)((char*)d_ws + 64); // 176 u64

    init_ws<<<1, 256, 0, stream>>>(sums, amax);
    main_pass<<<NPLANES * CHUNKS, TPB, 0, stream>>>(
        heat_p, offy_p, offx_p, heat_t, offy_t, offx_t, masks, sums, amax);
    final_pass<<<1, 256, 0, stream>>>(clss_p, clss_t, sums, amax, vw, dw,
                                      (float*)d_out);
}